// DGLHyperGINConv_27831388078170
// MI455X (gfx1250) — compile-verified
//
#include <hip/hip_runtime.h>

typedef __attribute__((ext_vector_type(2))) float v2f;
typedef __attribute__((ext_vector_type(8))) float v8f;

#define N_NODES  50000
#define N_HEDGES 25000
#define NNZ      800000
#define IN_CH    128
#define PROJ     128

// ---------------------------------------------------------------------------
// GEMM: Xp = X @ W  (fp32, V_WMMA_F32_16X16X4_F32), one wave per 16x16 tile.
// Also writes out = (1+eps) * Xp so the second scatter can accumulate into it.
// ---------------------------------------------------------------------------
__global__ __launch_bounds__(256) void gemm_wmma_f32(
    const float* __restrict__ X, const float* __restrict__ W,
    const float* __restrict__ eps, float* __restrict__ Xp,
    float* __restrict__ out)
{
    const int lane  = threadIdx.x & 31;
    const int wave  = threadIdx.x >> 5;
    const int tile  = blockIdx.x * 8 + wave;     // 25000 tiles total
    const int tileM = tile >> 3;                 // 3125 row tiles
    const int tileN = tile & 7;                  // 8 col tiles
    const int row0  = tileM * 16;
    const int col0  = tileN * 16;

    const int mrow  = lane & 15;                 // M (A) / N (B,C) index
    const int khalf = (lane >> 4) * 2;           // lanes 0-15 -> K 0,1 ; 16-31 -> K 2,3

    const float* __restrict__ Arow = X + (size_t)(row0 + mrow) * IN_CH;

    v8f c = {};
    #pragma unroll
    for (int k = 0; k < IN_CH; k += 4) {
        // A fragment: 16x4 f32, lane holds row mrow, K = k+khalf .. k+khalf+1 (b64 load)
        v2f a = *(const v2f*)(Arow + k + khalf);
        // B fragment: 4x16 f32, row-striped across lanes (mirrors A layout)
        v2f b;
        b.x = W[(size_t)(k + khalf)     * PROJ + col0 + mrow];
        b.y = W[(size_t)(k + khalf + 1) * PROJ + col0 + mrow];
        // (neg_a, A, neg_b, B, c_mod, C, reuse_a, reuse_b)
        c = __builtin_amdgcn_wmma_f32_16x16x4_f32(false, a, false, b,
                                                  (short)0, c, false, false);
    }

    const float scale = 1.0f + eps[0];
    #pragma unroll
    for (int i = 0; i < 8; ++i) {
        // C/D layout: VGPR i, lanes 0-15 -> M=i, lanes 16-31 -> M=i+8; N = lane%16
        const int m = row0 + i + 8 * (lane >> 4);
        const int n = col0 + (lane & 15);
        const float v = c[i];
        Xp[(size_t)m * PROJ + n]  = v;
        out[(size_t)m * PROJ + n] = scale * v;   // (1+eps)*Xp; Xv accumulated later
    }
}

// ---------------------------------------------------------------------------
// Zero the hyperedge accumulator (must run every call: harness doesn't re-zero).
// ---------------------------------------------------------------------------
__global__ void zero_f4(float4* __restrict__ p, int n4)
{
    int i = blockIdx.x * blockDim.x + threadIdx.x;
    if (i < n4) p[i] = make_float4(0.f, 0.f, 0.f, 0.f);
}

// ---------------------------------------------------------------------------
// Scatter-add: one wave per nonzero. Lane loads float4 (global_load_b128) from
// the gathered row and fires 4 native f32 atomics (no CAS loop) at the dest.
// ---------------------------------------------------------------------------
__device__ __forceinline__ void gatomic_add_f32(float* p, float v)
{
    asm volatile("global_atomic_add_f32 %0, %1, off"
                 : : "v"(p), "v"(v) : "memory");
}

__global__ __launch_bounds__(256) void scatter_add(
    const int* __restrict__ src, const int* __restrict__ dst,
    const float* __restrict__ in, float* __restrict__ acc, int nnz)
{
    const int e = blockIdx.x * 8 + (threadIdx.x >> 5);
    if (e >= nnz) return;
    const int lane = threadIdx.x & 31;

    const int s = src[e];
    const int d = dst[e];

    const float4 v = *(const float4*)(in + (size_t)s * PROJ + lane * 4);
    float* o = acc + (size_t)d * PROJ + lane * 4;
    gatomic_add_f32(o + 0, v.x);
    gatomic_add_f32(o + 1, v.y);
    gatomic_add_f32(o + 2, v.z);
    gatomic_add_f32(o + 3, v.w);
}

// ---------------------------------------------------------------------------
extern "C" void kernel_launch(void* const* d_in, const int* in_sizes, int n_in,
                              void* d_out, int out_size, void* d_ws, size_t ws_size,
                              hipStream_t stream)
{
    const float* X      = (const float*)d_in[0];
    const float* W      = (const float*)d_in[1];
    const float* eps    = (const float*)d_in[2];
    const int*   g1_src = (const int*)d_in[3];
    const int*   g1_dst = (const int*)d_in[4];
    const int*   g2_src = (const int*)d_in[5];
    const int*   g2_dst = (const int*)d_in[6];

    float* out = (float*)d_out;
    float* Xp  = (float*)d_ws;                         // 50000*128 f32 = 25.6 MB
    float* Xe  = Xp + (size_t)N_NODES * PROJ;          // 25000*128 f32 = 12.8 MB

    // 1) zero hyperedge accumulator
    const int n4 = N_HEDGES * PROJ / 4;
    zero_f4<<<(n4 + 255) / 256, 256, 0, stream>>>((float4*)Xe, n4);

    // 2) Xp = X @ W ; out = (1+eps)*Xp     (25000 tiles / 8 waves per block)
    gemm_wmma_f32<<<(N_NODES / 16) * (PROJ / 16) / 8, 256, 0, stream>>>(
        X, W, eps, Xp, out);

    // 3) Xe[g1_dst] += Xp[g1_src]
    scatter_add<<<NNZ / 8, 256, 0, stream>>>(g1_src, g1_dst, Xp, Xe, NNZ);

    // 4) out[g2_dst] += Xe[g2_src]   (out already holds (1+eps)*Xp)
    scatter_add<<<NNZ / 8, 256, 0, stream>>>(g2_src, g2_dst, Xe, out, NNZ);
}